// TimeWeightedCxrToEhrConv_22557168239252
// MI455X (gfx1250) — compile-verified
//
#include <hip/hip_runtime.h>

typedef float v2f __attribute__((ext_vector_type(2)));
typedef float v8f __attribute__((ext_vector_type(8)));

#define TAU_EPS_INV (1.0f / (0.5f + 1e-8f))

// Monotone order-preserving mapping float -> u32 (for atomicMax on floats).
__device__ __forceinline__ unsigned ord_key(float f) {
  unsigned u = __float_as_uint(f);
  return (u & 0x80000000u) ? ~u : (u | 0x80000000u);
}
__device__ __forceinline__ float ord_decode(unsigned o) {
  return (o & 0x80000000u) ? __uint_as_float(o & 0x7FFFFFFFu)
                           : __uint_as_float(~o);
}

// ---------------------------------------------------------------- init ----
__global__ void k_init(float* __restrict__ s_sum, unsigned* __restrict__ m_ord,
                       float* __restrict__ agg, int n_dst) {
  int i = blockIdx.x * blockDim.x + threadIdx.x;
  if (i < n_dst * 64) agg[i] = 0.0f;
  if (i < n_dst) {
    s_sum[i] = 0.0f;
    m_ord[i] = 0u;  // ord_key of any real float is > 0  => acts as -inf
  }
}

// ------------------------------------------------------- segment max ------
__global__ void k_segmax(const float* __restrict__ t, const int* __restrict__ dst,
                         unsigned* __restrict__ m_ord, int E) {
  int e = blockIdx.x * blockDim.x + threadIdx.x;
  if (e >= E) return;
  float score = t[e] * TAU_EPS_INV;
  atomicMax(&m_ord[dst[e]], ord_key(score));
}

// --------------------------------------------------- exp + segment sum ----
__global__ void k_expsum(const float* __restrict__ t, const int* __restrict__ dst,
                         const unsigned* __restrict__ m_ord,
                         float* __restrict__ ebuf, float* __restrict__ s_sum, int E) {
  int e = blockIdx.x * blockDim.x + threadIdx.x;
  if (e >= E) return;
  int d = dst[e];
  float score = t[e] * TAU_EPS_INV;
  float ev = __expf(score - ord_decode(m_ord[d]));
  ebuf[e] = ev;
  atomicAdd(&s_sum[d], ev);
}

// ------------------------------------- alpha-weighted gather/scatter ------
// One thread per (edge, 4 features): float4 gather of the x_src row, then 4
// f32 atomics into agg (25.6MB accumulator -> L2-resident atomic adds).
__global__ void k_scatter(const float* __restrict__ x_src,
                          const int* __restrict__ src, const int* __restrict__ dst,
                          const float* __restrict__ ebuf,
                          const float* __restrict__ s_sum,
                          float* __restrict__ agg, int E) {
  int tid = blockIdx.x * blockDim.x + threadIdx.x;
  int e = tid >> 4;
  if (e >= E) return;
  int h = (tid & 15) << 2;
  int d = dst[e];
  float alpha = ebuf[e] / (s_sum[d] + 1e-16f);
  const float4* xr = (const float4*)(x_src + src[e] * 64 + h);
  float4 v = *xr;
  float* a = agg + d * 64 + h;
  atomicAdd(a + 0, alpha * v.x);
  atomicAdd(a + 1, alpha * v.y);
  atomicAdd(a + 2, alpha * v.z);
  atomicAdd(a + 3, alpha * v.w);
}

// ------------------------------------------- out = agg @ W^T via WMMA -----
// out[d,j] = sum_k agg[d,k] * W[j,k].  A = agg tile 16x4 f32, B[k,j] = W[j,k].
// Block = 128 threads = 4 wave32s; wave w owns output columns [16w, 16w+16).
// All 16 B fragments preloaded per wave (W = 16KB, L0-resident); grid-stride
// over the 6250 M-tiles, 16 chained v_wmma_f32_16x16x4_f32 per tile.
__global__ void k_gemm_wmma(const float* __restrict__ agg,
                            const float* __restrict__ W,
                            float* __restrict__ out, int n_dst) {
  const int lane = threadIdx.x & 31;
  const int n0 = (threadIdx.x >> 5) << 4;   // wave id * 16
  const int col = lane & 15;
  const int kb = (lane >> 4) << 1;          // lanes 0-15: K={0,1}; 16-31: K={2,3}

  // B fragment layout mirrors A: VGPR0 = row K=kb, VGPR1 = row K=kb+1
  v2f bfrag[16];
#pragma unroll
  for (int kk = 0; kk < 16; ++kk) {
    const float* wp = W + (n0 + col) * 64 + kk * 4 + kb;
    bfrag[kk].x = wp[0];
    bfrag[kk].y = wp[1];
  }

  const int ntiles = (n_dst + 15) >> 4;
  for (int tm = blockIdx.x; tm < ntiles; tm += gridDim.x) {
    const int m0 = tm << 4;
    int rowA = m0 + (lane & 15);
    if (rowA >= n_dst) rowA = n_dst - 1;    // clamp loads on tail tile
    const float* ap = agg + rowA * 64 + kb;

    v8f c = {};
#pragma unroll
    for (int kk = 0; kk < 16; ++kk) {
      v2f a;
      a.x = ap[kk * 4 + 0];
      a.y = ap[kk * 4 + 1];
      c = __builtin_amdgcn_wmma_f32_16x16x4_f32(false, a, false, bfrag[kk],
                                                (short)0, c, false, false);
    }

    // D layout: VGPR r -> M = r (lanes 0-15) / M = 8+r (lanes 16-31)
    const int mbase = m0 + ((lane >> 4) << 3);
    const int cg = n0 + col;
    if (m0 + 16 <= n_dst) {
      // Uniform fast path: whole tile in range -> unconditional stores,
      // no per-row exec-mask manipulation.
#pragma unroll
      for (int r = 0; r < 8; ++r) {
        out[(mbase + r) * 64 + cg] = c[r];
      }
    } else {
      // Tail tile only: per-row guard.
#pragma unroll
      for (int r = 0; r < 8; ++r) {
        int rg = mbase + r;
        if (rg < n_dst) out[rg * 64 + cg] = c[r];
      }
    }
  }
}

// --------------------------------------------------------------- launch ---
extern "C" void kernel_launch(void* const* d_in, const int* in_sizes, int n_in,
                              void* d_out, int out_size, void* d_ws, size_t ws_size,
                              hipStream_t stream) {
  const float* x_src = (const float*)d_in[0];
  // d_in[1] (x_dst) only contributes its shape (num_dst)
  const float* W = (const float*)d_in[2];
  const int* edge_index = (const int*)d_in[3];   // [2, E] int32 (JAX x64 off)
  const float* t = (const float*)d_in[4];

  const int E = in_sizes[4];
  const int n_dst = in_sizes[1] / 64;
  const int* src = edge_index;
  const int* dst = edge_index + E;

  // Workspace carve-out (256B-aligned slices)
  char* ws = (char*)d_ws;
  size_t off = 0;
  auto bump = [&](size_t nbytes) {
    size_t cur = off;
    off += (nbytes + 255) & ~(size_t)255;
    return cur;
  };
  unsigned* m_ord = (unsigned*)(ws + bump((size_t)n_dst * 4));
  float* s_sum = (float*)(ws + bump((size_t)n_dst * 4));
  float* ebuf = (float*)(ws + bump((size_t)E * 4));
  float* agg = (float*)(ws + bump((size_t)n_dst * 64 * 4));
  (void)ws_size; (void)n_in; (void)out_size;

  const int B = 256;
  const int initN = n_dst * 64;
  k_init<<<(initN + B - 1) / B, B, 0, stream>>>(s_sum, m_ord, agg, n_dst);
  k_segmax<<<(E + B - 1) / B, B, 0, stream>>>(t, dst, m_ord, E);
  k_expsum<<<(E + B - 1) / B, B, 0, stream>>>(t, dst, m_ord, ebuf, s_sum, E);

  const long long scatterN = (long long)E * 16;
  k_scatter<<<(unsigned)((scatterN + B - 1) / B), B, 0, stream>>>(
      x_src, src, dst, ebuf, s_sum, agg, E);

  const int ntiles = (n_dst + 15) >> 4;
  const int gblocks = ntiles < 2048 ? ntiles : 2048;
  k_gemm_wmma<<<gblocks, 128, 0, stream>>>(agg, W, (float*)d_out, n_dst);
}